// FlexMaxPool_56891136803056
// MI455X (gfx1250) — compile-verified
//
#include <hip/hip_runtime.h>
#include <stdint.h>

// Problem constants from the reference: B=4, C=64, N=65536, K=16
#define B_   4
#define C_   64
#define N_   65536
#define K_   16
#define TILE 256   // n-values per block (8 wave32 waves)

__global__ __launch_bounds__(TILE) void flexmaxpool_kernel(
    const float* __restrict__ features,     // [B, C, N]
    const int*   __restrict__ neigh,        // [B, K, N]
    float*       __restrict__ out)          // [B, C, N]
{
    // LDS mirror of the index tile: 16 rows x 256 ints = 16 KB
    __shared__ int sidx[K_ * TILE];

    const int t    = threadIdx.x;
    const int tile = blockIdx.x;      // n-tile: 0..255
    const int b    = blockIdx.y;      // batch : 0..3
    const int n0   = tile * TILE;

    // ---------------------------------------------------------------
    // Stage neighborhood[b, 0:16, n0:n0+256] into LDS via the CDNA5
    // async global->LDS data mover (ASYNCcnt path). The tile is 1024
    // 16-byte chunks; each thread issues 4 b128 async loads.
    // Row k of the tile lives at nb + k*N (ints); its LDS mirror is
    // contiguous, so chunk c maps to LDS byte offset c*16 with
    // k = c/64, in-row byte offset = (c%64)*16.
    // ---------------------------------------------------------------
    const int* nb = neigh + (size_t)b * K_ * N_ + n0;   // uniform per block
    const unsigned lds_base = (unsigned)(uintptr_t)sidx;
#pragma unroll
    for (int i = 0; i < 4; ++i) {
        const int c     = i * TILE + t;                 // chunk id 0..1023
        const int k     = c >> 6;                       // row 0..15
        const unsigned voff =
            (unsigned)(k * (N_ * 4) + ((c & 63) << 4)); // global byte offset
        const unsigned ldso = lds_base + (unsigned)(c << 4);
        asm volatile("global_load_async_to_lds_b128 %0, %1, %2"
                     :
                     : "v"(ldso), "v"(voff), "s"(nb)
                     : "memory");
    }
    // Wait for this wave's async LDS writes, then sync the workgroup
    // (threads read chunks staged by other waves).
    asm volatile("s_wait_asynccnt 0x0" ::: "memory");
    __syncthreads();

    // Pull this thread's 16 indices into registers (bank-conflict-free:
    // lane L reads LDS word k*256 + L -> bank L%64, all distinct).
    int idx[K_];
#pragma unroll
    for (int k = 0; k < K_; ++k) idx[k] = sidx[k * TILE + t];

    // ---------------------------------------------------------------
    // Gather + max over K for all 64 channels, reusing the registers'
    // indices. Features are L2-resident (64 MiB << 192 MB L2); unroll
    // the channel loop x4 so ~64 independent gathers are in flight
    // (LOADcnt is 6 bits -> 63 outstanding loads is the per-wave cap).
    // ---------------------------------------------------------------
    const int    n  = n0 + t;
    const float* fb = features + (size_t)b * C_ * N_;
    float*       ob = out      + (size_t)b * C_ * N_ + n;

#pragma unroll 4
    for (int ch = 0; ch < C_; ++ch) {
        const float* fc = fb + (size_t)ch * N_;
        float m = fc[idx[0]];
#pragma unroll
        for (int k = 1; k < K_; ++k) {
            m = fmaxf(m, fc[idx[k]]);
        }
        ob[(size_t)ch * N_] = m;   // coalesced across lanes (contiguous n)
    }
}

extern "C" void kernel_launch(void* const* d_in, const int* in_sizes, int n_in,
                              void* d_out, int out_size, void* d_ws, size_t ws_size,
                              hipStream_t stream) {
    const float* features = (const float*)d_in[0];   // [B, C, N] f32
    const int*   neigh    = (const int*)  d_in[1];   // [B, K, N] i32
    float*       out      = (float*)d_out;           // [B, C, N] f32

    dim3 grid(N_ / TILE, B_);   // 256 x 4 blocks
    dim3 block(TILE);           // 256 threads = 8 wave32
    flexmaxpool_kernel<<<grid, block, 0, stream>>>(features, neigh, out);
}